// CriterionSFNet_7189775254113
// MI455X (gfx1250) — compile-verified
//
#include <hip/hip_runtime.h>
#include <hip/hip_bf16.h>
#include <math.h>

typedef float v2f __attribute__((ext_vector_type(2)));
typedef float v8f __attribute__((ext_vector_type(8)));

#define BATCH 4
#define NCLS 19
#define IH 64
#define IW 128
#define OH 512
#define OW 1024
#define PIX_PER_B (OH * OW)            // 524288 = 2^19
#define NPIX (BATCH * PIX_PER_B)       // 2097152
#define MIN_KEPT 100000
#define IGNORE_L 255

// ---- control block layout (32-bit words), placed after 4*NPIX floats in ws ----
#define C_HIST0   0     // 256 bins pred0
#define C_HIST1   256   // 256 bins pred1
#define C_PREFIX0 512
#define C_KREM0   513
#define C_PREFIX1 514
#define C_KREM1   515
#define C_SUM0    516   // f32 (atomic add)
#define C_CNT0    517   // u32
#define C_SUM1    518
#define C_CNT1    519
#define C_POS     520   // 4 f32
#define C_NEG     524   // 4 f32
#define C_BCE     528   // f32
#define C_WORDS   1024

// Cross-half lane shift: lane 16+i receives lane i's value (lower half: don't
// care, caller selects it away). Identity-select v_permlanex16_b32 — one VALU
// op, replaces ds_bpermute + s_wait_dscnt in the WMMA chain.
static __device__ __forceinline__ float xhalf_pull(float x) {
#if __has_builtin(__builtin_amdgcn_permlanex16)
    int r = __builtin_amdgcn_permlanex16(0, __float_as_int(x),
                                         0x76543210, (int)0xfedcba98,
                                         false, false);
    return __int_as_float(r);
#else
    return __shfl(x, (int)(threadIdx.x & 15), 32);
#endif
}

// ---------------------------------------------------------------------------
// init control block
// ---------------------------------------------------------------------------
__global__ void init_ctrl(unsigned* __restrict__ ctrl) {
    for (int i = threadIdx.x; i < C_WORDS; i += blockDim.x) ctrl[i] = 0u;
    __syncthreads();
    if (threadIdx.x == 0) {
        ctrl[C_KREM0] = MIN_KEPT;   // 1-based rank of the order statistic
        ctrl[C_KREM1] = MIN_KEPT;
    }
}

// ---------------------------------------------------------------------------
// Fused OHEM front-end: bilinear upsample (two chained V_WMMA_F32_16X16X4_F32
// per channel: O = Wy(16x4) @ X(4x4-in-16x4) @ Wx(4x16)) + online log-softmax.
// One wave computes one 16x16 output tile for all 19 classes.
// Writes per-pixel target-class prob (1.0 if ignored) and NLL (-1 if ignored).
// gridDim.z selects main(0)/coarse(1) prediction.
// ---------------------------------------------------------------------------
__global__ void ohem_fused(const float* __restrict__ mainp,
                           const float* __restrict__ coarsep,
                           const int*   __restrict__ seg,
                           float* __restrict__ wsf) {
    const int pred = blockIdx.z;
    const float* __restrict__ logits = pred ? coarsep : mainp;
    float* __restrict__ probv = wsf + (size_t)pred * 2u * NPIX;
    float* __restrict__ nllv  = probv + NPIX;

    const int lane = threadIdx.x & 31;
    const int wave = threadIdx.x >> 5;
    const int tile = blockIdx.x * 8 + wave;     // 8192 tiles total
    const int b  = tile >> 11;                  // / (32*64)
    const int tr = tile & 2047;
    const int ty = tr >> 6;                     // 0..31 (16-row tiles)
    const int tx = tr & 63;                     // 0..63 (16-col tiles)

    const float sy = 63.0f / 511.0f;            // align_corners scale
    const float sx = 127.0f / 1023.0f;

    int r0 = (int)((float)(ty * 16) * sy); if (r0 > IH - 4) r0 = IH - 4;
    int c0 = (int)((float)(tx * 16) * sx); if (c0 > IW - 4) c0 = IW - 4;

    const int M      = lane & 15;
    const int hiHalf = lane >> 4;               // 0 or 1
    const int k0 = hiHalf ? 2 : 0;              // K index held in VGPR0 slot
    const int k1 = hiHalf ? 3 : 1;              // K index held in VGPR1 slot

    // A2 = Wy (16x4) in WMMA A-layout: lane holds (M, k0) and (M, k1)
    float fy  = (float)(ty * 16 + M) * sy;
    int   iy0 = (int)fy; float wy = fy - (float)iy0;
    int   iy1 = min(iy0 + 1, IH - 1);
    int   a0 = iy0 - r0, a1 = iy1 - r0;
    v2f wyA;
    wyA.x = (a0 == k0 ? 1.0f - wy : 0.0f) + (a1 == k0 ? wy : 0.0f);
    wyA.y = (a0 == k1 ? 1.0f - wy : 0.0f) + (a1 == k1 ? wy : 0.0f);

    // B1 = Wx (4x16) in WMMA B-layout: lane holds (k0, N=M) and (k1, N=M)
    float fx  = (float)(tx * 16 + M) * sx;
    int   ix0 = (int)fx; float wxw = fx - (float)ix0;
    int   ix1 = min(ix0 + 1, IW - 1);
    int   b0i = ix0 - c0, b1i = ix1 - c0;
    v2f wxB;
    wxB.x = (b0i == k0 ? 1.0f - wxw : 0.0f) + (b1i == k0 ? wxw : 0.0f);
    wxB.y = (b0i == k1 ? 1.0f - wxw : 0.0f) + (b1i == k1 ? wxw : 0.0f);

    // This lane's 8 output pixels (D-layout): row = v + 8*hiHalf, col = M
    int tv[8];
    size_t obase = (size_t)b * PIX_PER_B + (size_t)(ty * 16) * OW + (size_t)(tx * 16);
#pragma unroll
    for (int v = 0; v < 8; ++v)
        tv[v] = seg[obase + (size_t)(v + 8 * hiHalf) * OW + M];

    float m8[8], s8[8], lt8[8];
#pragma unroll
    for (int v = 0; v < 8; ++v) { m8[v] = -1e30f; s8[v] = 0.0f; lt8[v] = 0.0f; }

    const float* xbase = logits + (size_t)b * NCLS * (IH * IW);
    const bool aLoad = (M < 4);
    const float* xrow = xbase + (size_t)(r0 + M) * IW + c0;  // only deref if M<4
    const v8f zeroC = {};

    for (int c = 0; c < NCLS; ++c) {
        // A1 = Xpad(16x4): rows 0..3 = input rows r0..r0+3, rest zero.
        // k1 == k0+1, so the two taps are one contiguous 8-byte load.
        v2f xa; xa.x = 0.0f; xa.y = 0.0f;
        if (aLoad) {
            const float* p = xrow + (size_t)c * (IH * IW);
            xa.x = p[k0];
            xa.y = p[k1];
            if (c + 1 < NCLS)
                __builtin_prefetch(xrow + (size_t)(c + 1) * (IH * IW), 0, 0);
        }
        // H(16x16) = Xpad @ Wx  : rows 0..3 are horizontally-interp'd input rows
        v8f Hd = __builtin_amdgcn_wmma_f32_16x16x4_f32(
            false, xa, false, wxB, (short)0, zeroC, false, false);

        // Repack H rows 0..3 (D-layout, lanes 0..15 of VGPR0..3) into B-layout:
        // upper half needs rows 2/3 pulled across from the lower half.
        float h2 = xhalf_pull(Hd[2]);
        float h3 = xhalf_pull(Hd[3]);
        v2f hb;
        hb.x = hiHalf ? h2 : Hd[0];
        hb.y = hiHalf ? h3 : Hd[1];

        // O(16x16) = Wy @ Hsub : the fully interpolated logit tile
        v8f O = __builtin_amdgcn_wmma_f32_16x16x4_f32(
            false, wyA, false, hb, (short)0, zeroC, false, false);

        // online softmax state update + target-logit capture
#pragma unroll
        for (int v = 0; v < 8; ++v) {
            float o  = O[v];
            float mn = fmaxf(m8[v], o);
            s8[v] = s8[v] * __expf(m8[v] - mn) + __expf(o - mn);
            m8[v] = mn;
            lt8[v] = (c == tv[v]) ? o : lt8[v];
        }
    }

#pragma unroll
    for (int v = 0; v < 8; ++v) {
        float lse   = m8[v] + __logf(s8[v]);
        float logp  = lt8[v] - lse;
        bool  valid = (tv[v] != IGNORE_L);
        size_t idx  = obase + (size_t)(v + 8 * hiHalf) * OW + M;
        probv[idx] = valid ? __expf(logp) : 1.0f;
        nllv[idx]  = valid ? -logp : -1.0f;
    }
}

// ---------------------------------------------------------------------------
// Radix select of the MIN_KEPT-th smallest prob (bit pattern of non-negative
// floats is order preserving). 4 passes of 8 bits, MSB first.
// ---------------------------------------------------------------------------
__global__ void sel_hist(const float* __restrict__ wsf, unsigned* __restrict__ ctrl,
                         int pass) {
    const int pred = blockIdx.y;
    const float* prob = wsf + (size_t)pred * 2u * NPIX;
    unsigned* hist = ctrl + (pred ? C_HIST1 : C_HIST0);
    const unsigned prefix = ctrl[pred ? C_PREFIX1 : C_PREFIX0];

    __shared__ unsigned lh[256];
    if (threadIdx.x < 256) lh[threadIdx.x] = 0u;
    __syncthreads();

    const int shift  = 24 - 8 * pass;
    const int stride = gridDim.x * blockDim.x;
    for (int i = blockIdx.x * blockDim.x + threadIdx.x; i < NPIX; i += stride) {
        unsigned v = __float_as_uint(prob[i]);
        bool part = (pass == 0) || ((v >> (shift + 8)) == (prefix >> (shift + 8)));
        if (part) atomicAdd(&lh[(v >> shift) & 255u], 1u);
    }
    __syncthreads();
    if (threadIdx.x < 256 && lh[threadIdx.x])
        atomicAdd(&hist[threadIdx.x], lh[threadIdx.x]);
}

__global__ void sel_scan(unsigned* __restrict__ ctrl, int pass) {
    const int pred = blockIdx.x;
    unsigned* hist = ctrl + (pred ? C_HIST1 : C_HIST0);
    const int t = threadIdx.x;

    __shared__ unsigned h[256];
    h[t] = hist[t];
    hist[t] = 0u;                    // re-zero for the next pass
    __syncthreads();
    for (int off = 1; off < 256; off <<= 1) {   // inclusive scan
        unsigned add = (t >= off) ? h[t - off] : 0u;
        __syncthreads();
        h[t] += add;
        __syncthreads();
    }
    const unsigned krem = ctrl[pred ? C_KREM1 : C_KREM0];
    const unsigned cum  = h[t];
    const unsigned prev = t ? h[t - 1] : 0u;
    if (cum >= krem && prev < krem) {           // single winner thread
        const int shift = 24 - 8 * pass;
        const int pi = pred ? C_PREFIX1 : C_PREFIX0;
        ctrl[pi] = ctrl[pi] | ((unsigned)t << shift);
        ctrl[pred ? C_KREM1 : C_KREM0] = krem - prev;
    }
}

// ---------------------------------------------------------------------------
// Masked NLL reduction with thr = max(0.7, kth-smallest prob)
// ---------------------------------------------------------------------------
__global__ void ohem_reduce(const float* __restrict__ wsf, unsigned* __restrict__ ctrl) {
    const int pred = blockIdx.y;
    const float* prob = wsf + (size_t)pred * 2u * NPIX;
    const float* nll  = prob + NPIX;
    const float thr = fmaxf(0.7f, __uint_as_float(ctrl[pred ? C_PREFIX1 : C_PREFIX0]));

    float ls = 0.0f; unsigned lc = 0u;
    const int stride = gridDim.x * blockDim.x;
    for (int i = blockIdx.x * blockDim.x + threadIdx.x; i < NPIX; i += stride) {
        float p = prob[i], nl = nll[i];
        if (nl >= 0.0f && p <= thr) { ls += nl; lc++; }
    }
    __shared__ float sf[256]; __shared__ unsigned su[256];
    sf[threadIdx.x] = ls; su[threadIdx.x] = lc;
    __syncthreads();
    for (int off = 128; off > 0; off >>= 1) {
        if (threadIdx.x < off) {
            sf[threadIdx.x] += sf[threadIdx.x + off];
            su[threadIdx.x] += su[threadIdx.x + off];
        }
        __syncthreads();
    }
    if (threadIdx.x == 0) {
        atomicAdd((float*)&ctrl[pred ? C_SUM1 : C_SUM0], sf[0]);
        atomicAdd(&ctrl[pred ? C_CNT1 : C_CNT0], su[0]);
    }
}

// ---------------------------------------------------------------------------
// Boundary BCE: per-sample pos/neg counts, then fused upsample + weighted BCE
// ---------------------------------------------------------------------------
__global__ void bce_counts(const float* __restrict__ bgt, unsigned* __restrict__ ctrl) {
    const int b = blockIdx.y;
    const float* t = bgt + (size_t)b * PIX_PER_B;
    unsigned lp = 0u, ln = 0u;
    const int stride = gridDim.x * blockDim.x;
    for (int i = blockIdx.x * blockDim.x + threadIdx.x; i < PIX_PER_B; i += stride) {
        float v = t[i];
        lp += (v == 1.0f); ln += (v == 0.0f);
    }
    __shared__ unsigned sp[256], sn[256];
    sp[threadIdx.x] = lp; sn[threadIdx.x] = ln;
    __syncthreads();
    for (int off = 128; off > 0; off >>= 1) {
        if (threadIdx.x < off) {
            sp[threadIdx.x] += sp[threadIdx.x + off];
            sn[threadIdx.x] += sn[threadIdx.x + off];
        }
        __syncthreads();
    }
    if (threadIdx.x == 0) {
        atomicAdd((float*)&ctrl[C_POS + b], (float)sp[0]);
        atomicAdd((float*)&ctrl[C_NEG + b], (float)sn[0]);
    }
}

__global__ void bce_main(const float* __restrict__ bp, const float* __restrict__ bgt,
                         unsigned* __restrict__ ctrl) {
    const float* cf = (const float*)ctrl;
    const float sy = 63.0f / 511.0f, sx = 127.0f / 1023.0f;
    float acc = 0.0f;
    const int stride = gridDim.x * blockDim.x;
    for (int i = blockIdx.x * blockDim.x + threadIdx.x; i < NPIX; i += stride) {
        int b  = i >> 19;
        int r  = i & (PIX_PER_B - 1);
        int oy = r >> 10, ox = r & 1023;
        float fy = (float)oy * sy, fx = (float)ox * sx;
        int iy0 = (int)fy, ix0 = (int)fx;
        float wy = fy - (float)iy0, wx = fx - (float)ix0;
        int iy1 = min(iy0 + 1, IH - 1), ix1 = min(ix0 + 1, IW - 1);
        const float* base = bp + (size_t)b * IH * IW;
        float v00 = base[iy0 * IW + ix0], v01 = base[iy0 * IW + ix1];
        float v10 = base[iy1 * IW + ix0], v11 = base[iy1 * IW + ix1];
        float p = (1.0f - wy) * ((1.0f - wx) * v00 + wx * v01)
                + wy * ((1.0f - wx) * v10 + wx * v11);
        float t = bgt[i];
        float pos = cf[C_POS + b], neg = cf[C_NEG + b];
        float valid = pos + neg;
        float wgt = (t == 1.0f) ? neg / valid : ((t == 0.0f) ? pos / valid : 0.0f);
        float lp  = fmaxf(__logf(p), -100.0f);
        float l1p = fmaxf(__logf(1.0f - p), -100.0f);
        acc += wgt * (-(t * lp + (1.0f - t) * l1p));
    }
    __shared__ float sf[256];
    sf[threadIdx.x] = acc;
    __syncthreads();
    for (int off = 128; off > 0; off >>= 1) {
        if (threadIdx.x < off) sf[threadIdx.x] += sf[threadIdx.x + off];
        __syncthreads();
    }
    if (threadIdx.x == 0) atomicAdd((float*)&ctrl[C_BCE], sf[0]);
}

__global__ void finalize(const unsigned* __restrict__ ctrl, float* __restrict__ out) {
    const float* cf = (const float*)ctrl;
    unsigned c0 = ctrl[C_CNT0]; if (c0 < 1u) c0 = 1u;
    unsigned c1 = ctrl[C_CNT1]; if (c1 < 1u) c1 = 1u;
    float l0 = cf[C_SUM0] / (float)c0;
    float l1 = cf[C_SUM1] / (float)c1;
    float lb = cf[C_BCE] / (float)NPIX;
    out[0] = l0 + l1 + lb;
}

// ---------------------------------------------------------------------------
extern "C" void kernel_launch(void* const* d_in, const int* in_sizes, int n_in,
                              void* d_out, int out_size, void* d_ws, size_t ws_size,
                              hipStream_t stream) {
    (void)in_sizes; (void)n_in; (void)out_size;
    const size_t needed = (size_t)4 * NPIX * sizeof(float) + C_WORDS * sizeof(unsigned);
    if (ws_size < needed) return;   // deterministic no-op if scratch is too small

    const float* mainp   = (const float*)d_in[0];
    const float* coarsep = (const float*)d_in[1];
    const float* bp      = (const float*)d_in[2];
    const int*   seg     = (const int*)d_in[3];
    const float* bgt     = (const float*)d_in[4];
    float*    wsf  = (float*)d_ws;
    unsigned* ctrl = (unsigned*)(wsf + (size_t)4 * NPIX);
    float* out = (float*)d_out;

    init_ctrl<<<1, 256, 0, stream>>>(ctrl);
    // 8192 tiles (16x16 out) x 2 preds, 8 waves/block, 1 wave = 1 tile
    ohem_fused<<<dim3(1024, 1, 2), 256, 0, stream>>>(mainp, coarsep, seg, wsf);
    for (int pass = 0; pass < 4; ++pass) {
        sel_hist<<<dim3(256, 2), 256, 0, stream>>>(wsf, ctrl, pass);
        sel_scan<<<2, 256, 0, stream>>>(ctrl, pass);
    }
    ohem_reduce<<<dim3(256, 2), 256, 0, stream>>>(wsf, ctrl);
    bce_counts<<<dim3(128, 4), 256, 0, stream>>>(bgt, ctrl);
    bce_main<<<512, 256, 0, stream>>>(bp, bgt, ctrl);
    finalize<<<1, 1, 0, stream>>>(ctrl, out);
}